// StrGNN_36902359007813
// MI455X (gfx1250) — compile-verified
//
#include <hip/hip_runtime.h>
#include <math.h>

#define T_SNAP 6
#define NN     20000
#define EE     400000
#define FF     64
#define CC     64
#define HH     128
#define K1     68      // F+3 = 67 padded to multiple of 4
#define KHOP   2

typedef __attribute__((ext_vector_type(2))) float v2f;
typedef __attribute__((ext_vector_type(8))) float v8f;

// ---------------- utility kernels ----------------

__global__ void k_zero_f(float* p, int n) {
    int i = blockIdx.x * blockDim.x + threadIdx.x;
    if (i < n) p[i] = 0.0f;
}

__global__ void k_zero_i(int* p, int n) {
    int i = blockIdx.x * blockDim.x + threadIdx.x;
    if (i < n) p[i] = 0;
}

__global__ void k_scatter_targets(const int* __restrict__ tgt, int* in_sub, int* isc) {
    int i = threadIdx.x;
    if (i < CC) {
        int n = tgt[i];
        in_sub[n] = 1;
        isc[n] = 1;
    }
}

// ---------------- BFS (k-hop expansion) ----------------

__global__ void k_bfs_edge(const int* __restrict__ src, const int* __restrict__ dst,
                           const int* __restrict__ in_sub, int* nxt) {
    int e = blockIdx.x * blockDim.x + threadIdx.x;
    if (e >= EE) return;
    int s = src[e], d = dst[e];
    if (in_sub[s]) nxt[d] = 1;   // benign race: all writers store 1
    if (in_sub[d]) nxt[s] = 1;
}

__global__ void k_bfs_merge(int* in_sub, int* nxt, int n) {
    int i = blockIdx.x * blockDim.x + threadIdx.x;
    if (i >= n) return;
    if (nxt[i]) in_sub[i] = 1;
    nxt[i] = 0;  // ready for next hop
}

// ---------------- GCN normalization ----------------

__global__ void k_deg_edge(const int* __restrict__ src, const int* __restrict__ dst,
                           const int* __restrict__ in_sub, float* deg) {
    int e = blockIdx.x * blockDim.x + threadIdx.x;
    if (e >= EE) return;
    int s = src[e], d = dst[e];
    if (in_sub[s] && in_sub[d]) unsafeAtomicAdd(&deg[d], 1.0f);
}

__global__ void k_dinv(const float* __restrict__ deg, const int* __restrict__ in_sub,
                       float* dinv, int n) {
    int i = blockIdx.x * blockDim.x + threadIdx.x;
    if (i >= n) return;
    float dg = deg[i] + (in_sub[i] ? 1.0f : 0.0f);
    dinv[i] = (dg > 0.0f) ? (1.0f / sqrtf(dg)) : 0.0f;
}

__global__ void k_enorm(const int* __restrict__ src, const int* __restrict__ dst,
                        const int* __restrict__ in_sub, const float* __restrict__ dinv,
                        float* enorm) {
    int e = blockIdx.x * blockDim.x + threadIdx.x;
    if (e >= EE) return;
    int s = src[e], d = dst[e];
    enorm[e] = (in_sub[s] && in_sub[d]) ? dinv[s] * dinv[d] : 0.0f;
}

// ---------------- feature assembly [N, K1] (x | labels | zero-pad) ----------------

__global__ void k_build_xl(const float* __restrict__ xt, const int* __restrict__ in_sub,
                           const int* __restrict__ isc, float* __restrict__ A) {
    int i = blockIdx.x * blockDim.x + threadIdx.x;
    if (i >= NN * K1) return;
    int n = i / K1;
    int c = i - n * K1;
    float v;
    if (c < FF)           v = xt[n * FF + c];
    else if (c == FF)     v = isc[n] ? 1.0f : 0.0f;                 // center label
    else if (c == FF + 1) v = (in_sub[n] && !isc[n]) ? 1.0f : 0.0f; // other subgraph
    else                  v = 0.0f;                                  // zero label + pad
    A[i] = v;
}

// ---------------- fp32 WMMA GEMM:  C[N,HH] = A[N,K] @ B[K,HH] ----------------
// Block: 256 threads = 8 wave32s. Each block covers 16 rows; wave w covers cols [16w,16w+16).
// V_WMMA_F32_16X16X4_F32 fragment layouts per CDNA5 ISA §7.12.2:
//   A 16x4: lanes 0-15 hold row M=lane (V0=K0,V1=K1); lanes 16-31 same rows (V0=K2,V1=K3)
//   B 4x16: lanes 0-15 hold col N=lane (V0=K0,V1=K1); lanes 16-31 (V0=K2,V1=K3)
//   C/D:    VGPR r = row r (lanes 0-15) / row r+8 (lanes 16-31), col = lane%16

__global__ void k_gemm(const float* __restrict__ A, int lda, int K,
                       const float* __restrict__ B, float* __restrict__ C) {
#if defined(__HIP_DEVICE_COMPILE__) && __has_builtin(__builtin_amdgcn_wmma_f32_16x16x4_f32)
    int wave  = threadIdx.x >> 5;
    int lane  = threadIdx.x & 31;
    int lhalf = lane >> 4;     // 0 or 1
    int lmod  = lane & 15;
    int rowBase = blockIdx.x * 16;
    int colBase = wave * 16;

    const float* arow = A + (long)(rowBase + lmod) * lda;
    v8f c = {0.f, 0.f, 0.f, 0.f, 0.f, 0.f, 0.f, 0.f};

    for (int k = 0; k < K; k += 4) {
        int kb = k + lhalf * 2;
        const float2 av = *(const float2*)(arow + kb);   // 8B aligned: lda even, kb even
        v2f a; a.x = av.x; a.y = av.y;
        v2f b;
        b.x = B[(kb    ) * HH + colBase + lmod];
        b.y = B[(kb + 1) * HH + colBase + lmod];
        c = __builtin_amdgcn_wmma_f32_16x16x4_f32(false, a, false, b, (short)0, c,
                                                  false, false);
    }

    float* out = C + (long)(rowBase + lhalf * 8) * HH + colBase + lmod;
#pragma unroll
    for (int r = 0; r < 8; ++r) out[r * HH] = c[r];
#else
    // scalar fallback (host pass / missing builtin): same tile decomposition
    int tid = threadIdx.x;
    for (int o = tid; o < 16 * HH; o += 256) {
        int r  = o / HH;
        int cc = o - r * HH;
        int row = blockIdx.x * 16 + r;
        float acc = 0.0f;
        for (int k = 0; k < K; ++k) acc += A[(long)row * lda + k] * B[k * HH + cc];
        C[(long)row * HH + cc] = acc;
    }
#endif
}

// ---------------- edge scatter-add: agg[dst] += hw[src] * enorm ----------------
// 32 lanes per edge, 4 channels per lane (float4 load = global_load_b128).

__global__ void k_edge_agg(const int* __restrict__ src, const int* __restrict__ dst,
                           const float* __restrict__ enorm, const float* __restrict__ hw,
                           float* agg) {
    int g = blockIdx.x * blockDim.x + threadIdx.x;
    int e = g >> 5;
    if (e >= EE) return;
    int lane = g & 31;
    float en = enorm[e];
    if (en == 0.0f) return;
    int s = src[e], d = dst[e];
    const float4 v = *(const float4*)(hw + (long)s * HH + lane * 4);
    float* o = agg + (long)d * HH + lane * 4;
    unsafeAtomicAdd(o + 0, v.x * en);
    unsafeAtomicAdd(o + 1, v.y * en);
    unsafeAtomicAdd(o + 2, v.z * en);
    unsafeAtomicAdd(o + 3, v.w * en);
}

// ---------------- out = relu(agg + hw*self_norm + bias) ----------------

__global__ void k_finalize(const float* __restrict__ agg, const float* __restrict__ hw,
                           const float* __restrict__ dinv, const int* __restrict__ in_sub,
                           const float* __restrict__ bias, float* __restrict__ out) {
    int i = blockIdx.x * blockDim.x + threadIdx.x;
    if (i >= NN * HH) return;
    int n = i >> 7;
    int c = i & (HH - 1);
    float sn = in_sub[n] ? dinv[n] * dinv[n] : 0.0f;
    float v = agg[i] + hw[i] * sn + bias[c];
    out[i] = v > 0.0f ? v : 0.0f;
}

// ---------------- argmax over last channel (first-occurrence tie-break) ----------------

__global__ void k_argmax_part(const float* __restrict__ h2, const int* __restrict__ in_sub,
                              float* pval, int* pidx) {
    __shared__ float sv[256];
    __shared__ int   si[256];
    int tid = threadIdx.x;
    float best = -INFINITY;
    int   bi   = 0x7fffffff;
    for (int n = blockIdx.x * 256 + tid; n < NN; n += gridDim.x * 256) {
        float v = in_sub[n] ? h2[(long)n * HH + (HH - 1)] : -INFINITY;
        if (v > best || (v == best && n < bi)) { best = v; bi = n; }
    }
    sv[tid] = best; si[tid] = bi;
    __syncthreads();
    for (int st = 128; st > 0; st >>= 1) {
        if (tid < st) {
            if (sv[tid + st] > sv[tid] ||
                (sv[tid + st] == sv[tid] && si[tid + st] < si[tid])) {
                sv[tid] = sv[tid + st]; si[tid] = si[tid + st];
            }
        }
        __syncthreads();
    }
    if (tid == 0) { pval[blockIdx.x] = sv[0]; pidx[blockIdx.x] = si[0]; }
}

__global__ void k_argmax_final(const float* __restrict__ pval, const int* __restrict__ pidx,
                               int nparts, const float* __restrict__ h2,
                               float* __restrict__ seq_t) {
    __shared__ float sv[64];
    __shared__ int   si[64];
    __shared__ int   sbest;
    int tid = threadIdx.x;  // 128 threads
    if (tid < 64) {
        if (tid < nparts) { sv[tid] = pval[tid]; si[tid] = pidx[tid]; }
        else              { sv[tid] = -INFINITY; si[tid] = 0x7fffffff; }
    }
    __syncthreads();
    for (int st = 32; st > 0; st >>= 1) {
        if (tid < st) {
            if (sv[tid + st] > sv[tid] ||
                (sv[tid + st] == sv[tid] && si[tid + st] < si[tid])) {
                sv[tid] = sv[tid + st]; si[tid] = si[tid + st];
            }
        }
        __syncthreads();
    }
    if (tid == 0) sbest = si[0];
    __syncthreads();
    seq_t[tid] = h2[(long)sbest * HH + tid];
}

// ---------------- pad W1 [67,128] -> [68,128] with a zero row ----------------

__global__ void k_w1pad(const float* __restrict__ W1, float* __restrict__ W1p) {
    int i = blockIdx.x * blockDim.x + threadIdx.x;
    if (i >= K1 * HH) return;
    int k = i / HH;
    W1p[i] = (k < FF + 3) ? W1[i] : 0.0f;
}

// ---------------- GRU over T snapshots + classifier head (1 block, 128 threads) -------

__global__ void k_gru_head(const float* __restrict__ seq,
                           const float* __restrict__ Wih, const float* __restrict__ Whh,
                           const float* __restrict__ bih, const float* __restrict__ bhh,
                           const float* __restrict__ Wc1, const float* __restrict__ bc1,
                           const float* __restrict__ Wc2, const float* __restrict__ bc2,
                           float* __restrict__ out) {
    __shared__ float sh[HH];
    __shared__ float sx[HH];
    __shared__ float shid[64];
    int i = threadIdx.x;
    sh[i] = 0.0f;
    __syncthreads();
    for (int t = 0; t < T_SNAP; ++t) {
        sx[i] = seq[t * HH + i];
        __syncthreads();
        float gir = bih[i], giz = bih[HH + i], gin = bih[2 * HH + i];
        float ghr = bhh[i], ghz = bhh[HH + i], ghn = bhh[2 * HH + i];
        for (int k = 0; k < HH; ++k) {
            float xv = sx[k], hv = sh[k];
            gir += xv * Wih[(0 * HH + i) * HH + k];
            giz += xv * Wih[(1 * HH + i) * HH + k];
            gin += xv * Wih[(2 * HH + i) * HH + k];
            ghr += hv * Whh[(0 * HH + i) * HH + k];
            ghz += hv * Whh[(1 * HH + i) * HH + k];
            ghn += hv * Whh[(2 * HH + i) * HH + k];
        }
        float r  = 1.0f / (1.0f + expf(-(gir + ghr)));
        float z  = 1.0f / (1.0f + expf(-(giz + ghz)));
        float nn = tanhf(gin + r * ghn);
        float hnew = (1.0f - z) * nn + z * sh[i];
        __syncthreads();
        sh[i] = hnew;
        __syncthreads();
    }
    if (i < 64) {
        float acc = bc1[i];
        for (int k = 0; k < HH; ++k) acc += sh[k] * Wc1[k * 64 + i];
        shid[i] = acc > 0.0f ? acc : 0.0f;
    }
    __syncthreads();
    if (i == 0) {
        float acc = bc2[0];
        for (int k = 0; k < 64; ++k) acc += shid[k] * Wc2[k];
        out[0] = 1.0f / (1.0f + expf(-acc));
    }
}

// ---------------- host orchestration ----------------

extern "C" void kernel_launch(void* const* d_in, const int* in_sizes, int n_in,
                              void* d_out, int out_size, void* d_ws, size_t ws_size,
                              hipStream_t stream) {
    const float* x   = (const float*)d_in[0];
    const int*   ei  = (const int*)  d_in[1];
    const int*   tgt = (const int*)  d_in[2];
    const float* W1  = (const float*)d_in[3];
    const float* b1  = (const float*)d_in[4];
    const float* W2  = (const float*)d_in[5];
    const float* b2  = (const float*)d_in[6];
    const float* Wih = (const float*)d_in[7];
    const float* Whh = (const float*)d_in[8];
    const float* bih = (const float*)d_in[9];
    const float* bhh = (const float*)d_in[10];
    const float* Wc1 = (const float*)d_in[11];
    const float* bc1 = (const float*)d_in[12];
    const float* Wc2 = (const float*)d_in[13];
    const float* bc2 = (const float*)d_in[14];
    float* out = (float*)d_out;

    // workspace carve-out (256B aligned regions)
    char* w = (char*)d_ws;
    auto carve = [&](size_t bytes) {
        char* p = w;
        w += (bytes + 255) & ~(size_t)255;
        return p;
    };
    int*   in_sub = (int*)  carve(sizeof(int)   * NN);
    int*   nxt    = (int*)  carve(sizeof(int)   * NN);
    int*   isc    = (int*)  carve(sizeof(int)   * NN);
    float* deg    = (float*)carve(sizeof(float) * NN);
    float* dinv   = (float*)carve(sizeof(float) * NN);
    float* enorm  = (float*)carve(sizeof(float) * EE);
    float* W1p    = (float*)carve(sizeof(float) * K1 * HH);
    float* A      = (float*)carve(sizeof(float) * (size_t)NN * K1);
    float* HW     = (float*)carve(sizeof(float) * (size_t)NN * HH);
    float* AGG    = (float*)carve(sizeof(float) * (size_t)NN * HH);
    float* H1     = (float*)carve(sizeof(float) * (size_t)NN * HH);
    float* H2     = (float*)carve(sizeof(float) * (size_t)NN * HH);
    float* seq    = (float*)carve(sizeof(float) * T_SNAP * HH);
    float* pval   = (float*)carve(sizeof(float) * 64);
    int*   pidx   = (int*)  carve(sizeof(int)   * 64);

    const int B = 256;
    const int gN   = (NN + B - 1) / B;
    const int gE   = (EE + B - 1) / B;
    const int gNH  = (NN * HH + B - 1) / B;
    const int gXL  = (NN * K1 + B - 1) / B;
    const int gEA  = (EE * 32) / B;           // 32 lanes per edge
    const int gGEM = NN / 16;                 // 16 rows per block (20000 % 16 == 0)

    k_w1pad<<<(K1 * HH + B - 1) / B, B, 0, stream>>>(W1, W1p);

    for (int t = 0; t < T_SNAP; ++t) {
        const float* xt  = x   + (size_t)t * NN * FF;
        const int*   src = ei  + (size_t)t * 2 * EE;
        const int*   dst = src + EE;
        const int*   tg  = tgt + (size_t)t * CC;

        // masks / degree init
        k_zero_i<<<gN, B, 0, stream>>>(in_sub, NN);
        k_zero_i<<<gN, B, 0, stream>>>(nxt, NN);
        k_zero_i<<<gN, B, 0, stream>>>(isc, NN);
        k_zero_f<<<gN, B, 0, stream>>>(deg, NN);
        k_scatter_targets<<<1, 64, 0, stream>>>(tg, in_sub, isc);

        // k-hop expansion
        for (int h = 0; h < KHOP; ++h) {
            k_bfs_edge<<<gE, B, 0, stream>>>(src, dst, in_sub, nxt);
            k_bfs_merge<<<gN, B, 0, stream>>>(in_sub, nxt, NN);
        }

        // GCN normalization
        k_deg_edge<<<gE, B, 0, stream>>>(src, dst, in_sub, deg);
        k_dinv<<<gN, B, 0, stream>>>(deg, in_sub, dinv, NN);
        k_enorm<<<gE, B, 0, stream>>>(src, dst, in_sub, dinv, enorm);

        // layer 1: xl @ W1 -> aggregate -> relu
        k_build_xl<<<gXL, B, 0, stream>>>(xt, in_sub, isc, A);
        k_gemm<<<gGEM, B, 0, stream>>>(A, K1, K1, W1p, HW);
        k_zero_f<<<gNH, B, 0, stream>>>(AGG, NN * HH);
        k_edge_agg<<<gEA, B, 0, stream>>>(src, dst, enorm, HW, AGG);
        k_finalize<<<gNH, B, 0, stream>>>(AGG, HW, dinv, in_sub, b1, H1);

        // layer 2: h1 @ W2 -> aggregate -> relu
        k_gemm<<<gGEM, B, 0, stream>>>(H1, HH, HH, W2, HW);
        k_zero_f<<<gNH, B, 0, stream>>>(AGG, NN * HH);
        k_edge_agg<<<gEA, B, 0, stream>>>(src, dst, enorm, HW, AGG);
        k_finalize<<<gNH, B, 0, stream>>>(AGG, HW, dinv, in_sub, b2, H2);

        // top-1 node by last channel -> seq[t]
        k_argmax_part<<<64, B, 0, stream>>>(H2, in_sub, pval, pidx);
        k_argmax_final<<<1, 128, 0, stream>>>(pval, pidx, 64, H2, seq + t * HH);
    }

    // GRU over snapshots + classifier head
    k_gru_head<<<1, 128, 0, stream>>>(seq, Wih, Whh, bih, bhh, Wc1, bc1, Wc2, bc2, out);
}